// DirectWarper_7035156431235
// MI455X (gfx1250) — compile-verified
//
#include <hip/hip_runtime.h>
#include <hip/hip_bf16.h>

// DirectWarper: forward depth warp (z-buffer splat) for B=8, H=480, W=640, C=3.
// Outputs: wd (B*H*W floats) then wc (B*C*H*W floats), concatenated in d_out.
//
// Primary path (needs ~19.7MB of d_ws): single-pass 64-bit atomic argmin.
//   key = (Zbits << 32) | (0xFFFFFFFF - pid); atomicMin(u64) implements the
//   reference's "min Z, tie-break max pid" total order exactly.
// Fallback path (zero-scratch beyond 384B): 3-pass u32 min / argmax scheme.

namespace {
constexpr int NB   = 8;
constexpr int HH   = 480;
constexpr int WW   = 640;
constexpr int NC   = 3;
constexpr int NPIX = HH * WW;          // 307200 = 1200 * 256
constexpr unsigned INF_BITS = 0x7F800000u;
constexpr unsigned long long EMPTY_KEY = ~0ULL;
}

typedef float    fv2  __attribute__((ext_vector_type(2)));
typedef float    fv4  __attribute__((ext_vector_type(4)));
typedef unsigned uv4  __attribute__((ext_vector_type(4)));

// ---------------------------------------------------------------------------
// Per-batch fused matrices:  rot = K @ P[:,:3] @ inv(K),  tr = K @ P[:,3]
// mats[b*12 + {0..8 rot row-major, 9..11 tr}]
// ---------------------------------------------------------------------------
__global__ void dw_mats_kernel(const float* __restrict__ intr,
                               const float* __restrict__ pose,
                               float* __restrict__ mats) {
    const int b = threadIdx.x;
    if (b >= NB) return;

    const float* K = intr + b * 9;   // row-major 3x3
    const float* P = pose + b * 12;  // row-major 3x4

    const float k00 = K[0], k01 = K[1], k02 = K[2];
    const float k10 = K[3], k11 = K[4], k12 = K[5];
    const float k20 = K[6], k21 = K[7], k22 = K[8];

    const float c00 =  (k11 * k22 - k12 * k21);
    const float c01 = -(k10 * k22 - k12 * k20);
    const float c02 =  (k10 * k21 - k11 * k20);
    const float c10 = -(k01 * k22 - k02 * k21);
    const float c11 =  (k00 * k22 - k02 * k20);
    const float c12 = -(k00 * k21 - k01 * k20);
    const float c20 =  (k01 * k12 - k02 * k11);
    const float c21 = -(k00 * k12 - k02 * k10);
    const float c22 =  (k00 * k11 - k01 * k10);
    const float det = k00 * c00 + k01 * c01 + k02 * c02;
    const float id  = 1.0f / det;

    float inv[9];
    inv[0] = c00 * id; inv[1] = c10 * id; inv[2] = c20 * id;
    inv[3] = c01 * id; inv[4] = c11 * id; inv[5] = c21 * id;
    inv[6] = c02 * id; inv[7] = c12 * id; inv[8] = c22 * id;

    float m1[9];
    #pragma unroll
    for (int r = 0; r < 3; ++r)
        #pragma unroll
        for (int c = 0; c < 3; ++c)
            m1[r * 3 + c] = K[r * 3 + 0] * P[0 * 4 + c]
                          + K[r * 3 + 1] * P[1 * 4 + c]
                          + K[r * 3 + 2] * P[2 * 4 + c];

    float* m = mats + b * 12;
    #pragma unroll
    for (int r = 0; r < 3; ++r)
        #pragma unroll
        for (int c = 0; c < 3; ++c)
            m[r * 3 + c] = m1[r * 3 + 0] * inv[0 * 3 + c]
                         + m1[r * 3 + 1] * inv[1 * 3 + c]
                         + m1[r * 3 + 2] * inv[2 * 3 + c];

    #pragma unroll
    for (int r = 0; r < 3; ++r)
        m[9 + r] = K[r * 3 + 0] * P[3] + K[r * 3 + 1] * P[7] + K[r * 3 + 2] * P[11];
}

// ---------------------------------------------------------------------------
// Shared projection core (i, j already split out).
// ---------------------------------------------------------------------------
__device__ __forceinline__ bool dw_project_ij(const float* __restrict__ m,
                                              float d, float jx, float iy,
                                              unsigned& zbits, int& flat) {
    const float xd = jx * d;
    const float yd = iy * d;

    const float tp0 = fmaf(m[0], xd, fmaf(m[1], yd, fmaf(m[2], d, m[9])));
    const float tp1 = fmaf(m[3], xd, fmaf(m[4], yd, fmaf(m[5], d, m[10])));
    const float Z   = fmaf(m[6], xd, fmaf(m[7], yd, fmaf(m[8], d, m[11])));

    const float Uf = rintf(tp0 / Z);   // rintf == round-half-to-even == jnp.round
    const float Vf = rintf(tp1 / Z);

    const bool valid = (Z > 0.001f) &&
                       (Uf >= 0.0f) && (Uf < (float)WW) &&
                       (Vf >= 0.0f) && (Vf < (float)HH);
    if (!valid) return false;
    flat  = (int)Vf * WW + (int)Uf;
    zbits = __float_as_uint(Z);
    return true;
}

// ===========================================================================
// PRIMARY PATH: single-pass 64-bit atomic argmin via key buffer in d_ws.
// ===========================================================================
__global__ void dw_init64_kernel(unsigned long long* __restrict__ keys) {
    const int b = blockIdx.y;
    const int t = blockIdx.x * blockDim.x + threadIdx.x;   // one thread = 2 keys
    if (t < NPIX / 2) {
        // Regular-temporal on purpose: these lines are immediately RMW'd by
        // L2 atomics and re-read by resolve — keep them resident.
        uv4* p = (uv4*)(keys + (size_t)b * NPIX) + t;      // b128 store
        *p = (uv4){~0u, ~0u, ~0u, ~0u};
    }
}

// 4 consecutive pixels per thread: one b128 depth load, one row split
// (W % 4 == 0 so all 4 pixels share the same row i).
__global__ void dw_splat64_kernel(const float* __restrict__ depth,
                                  const float* __restrict__ mats,
                                  unsigned long long* __restrict__ keys) {
    const int b = blockIdx.y;
    const int t = blockIdx.x * blockDim.x + threadIdx.x;
    if (t >= NPIX / 4) return;

    const int base = t * 4;
    const float* m = mats + b * 12;   // uniform per block -> scalar loads

    const fv4 d4 = *(const fv4*)(depth + (size_t)b * NPIX + base);

    const int   i  = base / WW;
    const int   j0 = base - i * WW;
    const float iy = (float)i;

    unsigned long long* kb = keys + (size_t)b * NPIX;

    #pragma unroll
    for (int k = 0; k < 4; ++k) {
        unsigned zbits; int flat;
        if (dw_project_ij(m, d4[k], (float)(j0 + k), iy, zbits, flat)) {
            // Total order: Z ascending, then pid descending -> one atomic does
            // both the z-min and the reference's max-pid tie-break.
            const unsigned long long key =
                ((unsigned long long)zbits << 32) |
                (unsigned long long)(0xFFFFFFFFu - (unsigned)(base + k));
            atomicMin(&kb[flat], key);                     // global_atomic_min_u64
        }
    }
}

// 2 pixels per thread: non-temporal b128 key load (keys are dead after this
// pass), regular-temporal color gathers (high intra-line reuse), and
// non-temporal b64 output stores (outputs never re-read on device).
__global__ void dw_resolve64_kernel(const unsigned long long* __restrict__ keys,
                                    const float* __restrict__ img,
                                    float* __restrict__ out) {
    const int b = blockIdx.y;
    const int t = blockIdx.x * blockDim.x + threadIdx.x;
    if (t >= NPIX / 2) return;

    const int pix0 = t * 2;

    const uv4 kk = __builtin_nontemporal_load(
        (const uv4*)(keys + (size_t)b * NPIX + pix0));
    const unsigned long long key0 = ((unsigned long long)kk.y << 32) | kk.x;
    const unsigned long long key1 = ((unsigned long long)kk.w << 32) | kk.z;

    const float* imgb = img + (size_t)b * (NC * NPIX);

    fv2 wdv = {0.0f, 0.0f};
    fv2 p0  = {0.0f, 0.0f};
    fv2 p1  = {0.0f, 0.0f};
    fv2 p2  = {0.0f, 0.0f};

    if (key0 != EMPTY_KEY) {
        wdv.x = __uint_as_float((unsigned)(key0 >> 32));
        const int idx = (int)(0xFFFFFFFFu - (unsigned)(key0 & 0xFFFFFFFFu));
        p0.x = imgb[idx];
        p1.x = imgb[NPIX + idx];
        p2.x = imgb[2 * NPIX + idx];
    }
    if (key1 != EMPTY_KEY) {
        wdv.y = __uint_as_float((unsigned)(key1 >> 32));
        const int idx = (int)(0xFFFFFFFFu - (unsigned)(key1 & 0xFFFFFFFFu));
        p0.y = imgb[idx];
        p1.y = imgb[NPIX + idx];
        p2.y = imgb[2 * NPIX + idx];
    }

    float* wd = out;
    float* wc = out + NB * NPIX;
    __builtin_nontemporal_store(wdv, (fv2*)(wd + (size_t)b * NPIX + pix0));
    __builtin_nontemporal_store(p0,  (fv2*)(wc + (size_t)b * (NC * NPIX) + pix0));
    __builtin_nontemporal_store(p1,  (fv2*)(wc + (size_t)b * (NC * NPIX) + NPIX + pix0));
    __builtin_nontemporal_store(p2,  (fv2*)(wc + (size_t)b * (NC * NPIX) + 2 * NPIX + pix0));
}

// ===========================================================================
// FALLBACK PATH (zero large scratch): 3-pass u32 min + argmax, buffers live
// inside d_out (wd region = z-buffer, wc c=0 plane = index plane).
// ===========================================================================
__global__ void dw_init_kernel(unsigned* __restrict__ wdbits,
                               int* __restrict__ wcplane) {
    const int b   = blockIdx.y;
    const int pix = blockIdx.x * blockDim.x + threadIdx.x;
    if (pix < NPIX) {
        wdbits[b * NPIX + pix]         = INF_BITS;
        wcplane[b * (NC * NPIX) + pix] = -1;
    }
}

__global__ void dw_splat_kernel(const float* __restrict__ depth,
                                const float* __restrict__ mats,
                                unsigned* __restrict__ wdbits,
                                int* __restrict__ idxplane,
                                int mode) {
    const int b   = blockIdx.y;
    const int pix = blockIdx.x * blockDim.x + threadIdx.x;
    if (pix >= NPIX) return;

    const float* m = mats + b * 12;
    const float  d = depth[b * NPIX + pix];
    const int    i = pix / WW;
    const int    j = pix - i * WW;

    unsigned zbits; int flat;
    if (!dw_project_ij(m, d, (float)j, (float)i, zbits, flat)) return;
    // Same compiled kernel for both modes -> identical Z bits.
    if (mode == 0) {
        atomicMin(&wdbits[b * NPIX + flat], zbits);
    } else if (wdbits[b * NPIX + flat] == zbits) {
        atomicMax(&idxplane[b * (NC * NPIX) + flat], pix);
    }
}

__global__ void dw_resolve_kernel(const float* __restrict__ img,
                                  float* __restrict__ out) {
    const int b   = blockIdx.y;
    const int pix = blockIdx.x * blockDim.x + threadIdx.x;
    if (pix >= NPIX) return;

    float* wd = out;
    float* wc = out + NB * NPIX;

    const unsigned bits = __float_as_uint(wd[b * NPIX + pix]);
    const int      idx  = ((const int*)wc)[b * (NC * NPIX) + pix];

    wd[b * NPIX + pix] = (bits == INF_BITS) ? 0.0f : __uint_as_float(bits);

    const float* imgb = img + (size_t)b * (NC * NPIX);
    float c0 = 0.0f, c1 = 0.0f, c2 = 0.0f;
    if (idx >= 0) {
        c0 = imgb[idx];
        c1 = imgb[NPIX + idx];
        c2 = imgb[2 * NPIX + idx];
    }
    wc[b * (NC * NPIX) + pix]            = c0;
    wc[b * (NC * NPIX) + NPIX + pix]     = c1;
    wc[b * (NC * NPIX) + 2 * NPIX + pix] = c2;
}

// ---------------------------------------------------------------------------
// Host launcher. Inputs: depth, img, pose_matrix, intrinsics, dilation(unused).
// ws layout: [0,384)   per-batch fused matrices
//            [512, 512 + B*NPIX*8)  64-bit key buffer (primary path only)
// The ws_size branch is constant across calls -> deterministic work.
// ---------------------------------------------------------------------------
extern "C" void kernel_launch(void* const* d_in, const int* in_sizes, int n_in,
                              void* d_out, int out_size, void* d_ws, size_t ws_size,
                              hipStream_t stream) {
    (void)in_sizes; (void)n_in; (void)out_size;

    const float* depth = (const float*)d_in[0];
    const float* img   = (const float*)d_in[1];
    const float* pose  = (const float*)d_in[2];
    const float* intr  = (const float*)d_in[3];
    // d_in[4] = dilation, unused by the reference.

    float* out  = (float*)d_out;
    float* mats = (float*)d_ws;

    const dim3 blk(256);
    const size_t keys_bytes = (size_t)NB * NPIX * sizeof(unsigned long long);

    dw_mats_kernel<<<1, 32, 0, stream>>>(intr, pose, mats);

    if (ws_size >= 512 + keys_bytes) {
        // -------- primary: single-pass 64-bit argmin --------
        unsigned long long* keys = (unsigned long long*)((char*)d_ws + 512);
        const dim3 grd_half(NPIX / 2 / 256, NB);   // 600 x 8  (2 elems/thread)
        const dim3 grd_quar(NPIX / 4 / 256, NB);   // 300 x 8  (4 elems/thread)
        dw_init64_kernel<<<grd_half, blk, 0, stream>>>(keys);
        dw_splat64_kernel<<<grd_quar, blk, 0, stream>>>(depth, mats, keys);
        dw_resolve64_kernel<<<grd_half, blk, 0, stream>>>(keys, img, out);
    } else {
        // -------- fallback: 3-pass u32 min / argmax in d_out --------
        const dim3 grd(NPIX / 256, NB);            // 1200 x 8
        unsigned* wdbits   = (unsigned*)out;
        int*      idxplane = (int*)(out + NB * NPIX);
        dw_init_kernel<<<grd, blk, 0, stream>>>(wdbits, idxplane);
        dw_splat_kernel<<<grd, blk, 0, stream>>>(depth, mats, wdbits, idxplane, 0);
        dw_splat_kernel<<<grd, blk, 0, stream>>>(depth, mats, wdbits, idxplane, 1);
        dw_resolve_kernel<<<grd, blk, 0, stream>>>(img, out);
    }
}